// CrossModalAttention_25383256719879
// MI455X (gfx1250) — compile-verified
//
#include <hip/hip_runtime.h>
#include <math.h>

typedef float v2f __attribute__((ext_vector_type(2)));
typedef float v8f __attribute__((ext_vector_type(8)));

#define BATCH 4
#define SEQ   197
#define DIM   768
#define HID   256
#define ROWS  (BATCH*SEQ)                 /* 788 */
#define PAIRS (BATCH*SEQ*SEQ)             /* 155236 */
#define TOT_F 38809.0f                    /* 197*197 */
#define KMAX  1536                        /* largest GEMM K; zero-row length */

// ---------------- wave helpers (wave32 on gfx1250) ----------------
__device__ __forceinline__ float wave_sum(float v){
#pragma unroll
  for (int o = 16; o > 0; o >>= 1) v += __shfl_xor(v, o, 32);
  return v;
}
__device__ __forceinline__ float wave_max(float v){
#pragma unroll
  for (int o = 16; o > 0; o >>= 1) v = fmaxf(v, __shfl_xor(v, o, 32));
  return v;
}

// ---------------- setup kernels ----------------
// zero the accumulators (16) and the shared zero-row (KMAX) used for OOB GEMM lanes
__global__ void init_ws_kernel(float* __restrict__ zrow, float* __restrict__ acc){
  int t = threadIdx.x;
  for (int i = t; i < KMAX; i += 256) zrow[i] = 0.f;
  if (t < 16) acc[t] = 0.f;
}

// img_global / txt_global: column means over SEQ
__global__ void globals_kernel(const float* __restrict__ img, const float* __restrict__ txt,
                               float* __restrict__ gimg, float* __restrict__ gtxt){
  int idx = blockIdx.x * 256 + threadIdx.x;            // 2*BATCH*DIM
  if (idx >= 2 * BATCH * DIM) return;
  int which = idx / (BATCH * DIM);
  int r = idx % (BATCH * DIM);
  int b = r / DIM, d = r % DIM;
  const float* X = which ? txt : img;
  float s = 0.f;
  for (int l = 0; l < SEQ; ++l) s += X[((long)b*SEQ + l)*DIM + d];
  (which ? gtxt : gimg)[r] = s * (1.0f/SEQ);
}

// X[r,0:768]=seq row, X[r,768:1536]=global(b)
__global__ void build_x_kernel(const float* __restrict__ seq, const float* __restrict__ g,
                               float* __restrict__ X){
  long idx = (long)blockIdx.x * 256 + threadIdx.x;     // ROWS*1536
  if (idx >= (long)ROWS * 1536) return;
  int r = (int)(idx / 1536), c = (int)(idx % 1536);
  int b = r / SEQ;
  X[idx] = (c < DIM) ? seq[(long)r*DIM + c] : g[b*DIM + (c - DIM)];
}

// Wa = [W1q | W1qg] (cols 0:768 ++ 1536:2304), Wb = [W1k | W1kg] (768:1536 ++ 2304:3072)
__global__ void build_w_kernel(const float* __restrict__ W1,
                               float* __restrict__ Wa, float* __restrict__ Wb){
  long idx = (long)blockIdx.x * 256 + threadIdx.x;     // HID*1536
  if (idx >= (long)HID * 1536) return;
  int n = (int)(idx / 1536), c = (int)(idx % 1536);
  long base = (long)n * (4*DIM);
  Wa[idx] = (c < DIM) ? W1[base + c]        : W1[base + 2*DIM + (c - DIM)];
  Wb[idx] = (c < DIM) ? W1[base + DIM + c]  : W1[base + 3*DIM + (c - DIM)];
}

// ---------------- fp32 WMMA GEMM:  C = alpha*(A @ B^T) + bias ----------------
// A: [M,K] lda, B: [N,K] ldb, C: [M,N] ldc. Batched via strides.
// One wave computes a 16 x (16*NW) tile with V_WMMA_F32_16X16X4_F32; the A fragment
// is reused across the NW B fragments. OOB lanes read from a zeroed scratch row so
// the K-loop has no predication (EXEC stays all-ones; loads are unconditional).
template<int NW>
__global__ void gemm_bt_wmma(const float* __restrict__ A, const float* __restrict__ Bw,
                             const float* __restrict__ bias, float* __restrict__ C,
                             const float* __restrict__ zrow,
                             int M, int N, int K, int lda, int ldb, int ldc,
                             long bsA, long bsB, long bsC,
                             int tilesM, int tilesN, float alpha)
{
  int wave = threadIdx.x >> 5;
  int lane = threadIdx.x & 31;
  int tile = blockIdx.x * 8 + wave;
  if (tile >= tilesM * tilesN) return;                 // wave-uniform exit
  int b  = blockIdx.y;
  int tm = tile / tilesN, tn = tile % tilesN;
  int l15 = lane & 15, half = lane >> 4;
  int koff = 2*half;                                   // lanes 16..31 hold K+2,K+3
  int ar = tm*16 + l15;
  bool aval = ar < M;
  const float* arow = (aval ? (A + bsA*(long)b + (long)ar*lda) : zrow) + koff;
  const float* brow[NW];
  bool bval[NW];
  int bn[NW];
#pragma unroll
  for (int j = 0; j < NW; ++j) {
    bn[j]   = (tn*NW + j)*16 + l15;
    bval[j] = bn[j] < N;
    brow[j] = (bval[j] ? (Bw + bsB*(long)b + (long)bn[j]*ldb) : zrow) + koff;
  }
  v8f acc[NW];
  v8f z8 = {};
#pragma unroll
  for (int j = 0; j < NW; ++j) acc[j] = z8;
#pragma unroll 4
  for (int k0 = 0; k0 < K; k0 += 4) {
    v2f a2 = *(const v2f*)(arow + k0);
#pragma unroll
    for (int j = 0; j < NW; ++j) {
      v2f b2 = *(const v2f*)(brow[j] + k0);
      acc[j] = __builtin_amdgcn_wmma_f32_16x16x4_f32(false, a2, false, b2,
                                                     (short)0, acc[j], false, false);
    }
  }
  float* crow = C + bsC*(long)b;
#pragma unroll
  for (int j = 0; j < NW; ++j) {
    float bv = (bias != nullptr && bval[j]) ? bias[bn[j]] : 0.f;
#pragma unroll
    for (int r = 0; r < 8; ++r) {
      int cm = tm*16 + r + 8*half;                     // C layout: vgpr r -> rows r / r+8
      if (cm < M && bval[j]) crow[(long)cm*ldc + bn[j]] = alpha*acc[j][r] + bv;
    }
  }
}

// ---------------- masker: gelu+LN(256)+W2 threshold, gumbel hard mask ----------------
// a: [ROWS,256] (b1 folded), bt: [ROWS,256], sim: [B,SEQ,SEQ], gn: [B,SEQ,SEQ,2]
__global__ void masker_kernel(const float* __restrict__ a, const float* __restrict__ bt,
                              const float* __restrict__ gamma, const float* __restrict__ beta,
                              const float* __restrict__ W2, const float* __restrict__ b2,
                              const float* __restrict__ sim, const float* __restrict__ gn,
                              float* __restrict__ mask, int simT)
{
  int wave = threadIdx.x >> 5, lane = threadIdx.x & 31;
  long pair = (long)blockIdx.x * 8 + wave;
  if (pair >= (long)PAIRS) return;
  int s = (int)(pair % SEQ);
  long t1 = pair / SEQ;
  int l = (int)(t1 % SEQ);
  int b = (int)(t1 / SEQ);
  const float* ar = a  + ((long)b*SEQ + l) * HID;
  const float* br = bt + ((long)b*SEQ + s) * HID;
  float h[8], s1 = 0.f, s2 = 0.f;
#pragma unroll
  for (int j = 0; j < 8; ++j) {
    int c = lane + 32*j;
    float x = ar[c] + br[c];
    x = 0.5f * x * (1.0f + erff(x * 0.70710678118654752f));   // exact gelu
    h[j] = x; s1 += x; s2 += x*x;
  }
  s1 = wave_sum(s1); s2 = wave_sum(s2);
  float mu  = s1 * (1.0f/HID);
  float var = s2 * (1.0f/HID) - mu*mu;
  float inv = 1.0f / sqrtf(var + 1e-5f);
  float thr = 0.f;
#pragma unroll
  for (int j = 0; j < 8; ++j) {
    int c = lane + 32*j;
    float hn = (h[j] - mu) * inv * gamma[c] + beta[c];
    thr += hn * W2[c];
  }
  thr = wave_sum(thr) + b2[0];
  if (lane == 0) {
    long ls = ((long)b*SEQ + l)*SEQ + s;
    float sv = simT ? sim[((long)b*SEQ + s)*SEQ + l] : sim[ls];
    float logit = (sv - thr) * 0.1f;                   // /TEMP(=10)
    float g0 = gn[ls*2], g1 = gn[ls*2 + 1];
    // straight-through gumbel: forward value == hard argmax (ties -> index 0)
    mask[ls] = ((logit + g1) > (-logit + g0)) ? 1.0f : 0.0f;
  }
}

// ---------------- masked multihead attention: one block per (b,l), wave per head ----------------
__global__ void attention_kernel(const float* __restrict__ q, const float* __restrict__ kv,
                                 const float* __restrict__ mask, float* __restrict__ outp)
{
  __shared__ float qs[8][96];
  __shared__ float attn[8][SEQ];
  __shared__ float mrow[SEQ];
  int bl = blockIdx.x;
  int b  = bl / SEQ;
  int head = threadIdx.x >> 5;
  int lane = threadIdx.x & 31;
  for (int s = threadIdx.x; s < SEQ; s += 256) mrow[s] = mask[(long)bl*SEQ + s];
  const float* qrow = q + (long)bl*DIM + head*96;
  for (int d = lane; d < 96; d += 32) qs[head][d] = qrow[d];
  __syncthreads();
  const float* kbase = kv + (long)b*SEQ*(2*DIM) + head*96;
  float sc[7];
  float ninf = -__builtin_inff();
#pragma unroll
  for (int j = 0; j < 7; ++j) {
    int s = lane + 32*j;
    float v = ninf;
    if (s < SEQ && mrow[s] < 0.5f) {
      const float* kr = kbase + (long)s*(2*DIM);
      float acc = 0.f;
      for (int d = 0; d < 96; ++d) acc += qs[head][d] * kr[d];
      v = acc * 0.10206207261596575f;                  // 1/sqrt(96)
    }
    sc[j] = v;
  }
  float mx = sc[0];
#pragma unroll
  for (int j = 1; j < 7; ++j) mx = fmaxf(mx, sc[j]);
  mx = wave_max(mx);
  if (mx == ninf) {                                    // fully masked row -> uniform
#pragma unroll
    for (int j = 0; j < 7; ++j) { int s = lane + 32*j; if (s < SEQ) attn[head][s] = 1.0f/SEQ; }
  } else {
    float e[7], ssum = 0.f;
#pragma unroll
    for (int j = 0; j < 7; ++j) { e[j] = (sc[j] == ninf) ? 0.f : expf(sc[j] - mx); ssum += e[j]; }
    ssum = wave_sum(ssum);
    float invs = 1.0f / ssum;
#pragma unroll
    for (int j = 0; j < 7; ++j) { int s = lane + 32*j; if (s < SEQ) attn[head][s] = e[j]*invs; }
  }
  __syncthreads();
  const float* vbase = kv + (long)b*SEQ*(2*DIM) + DIM + head*96;
  int d0 = lane * 3;                                   // 96 = 32 lanes * 3
  float o0 = 0.f, o1 = 0.f, o2 = 0.f;
  for (int s = 0; s < SEQ; ++s) {
    float aw = attn[head][s];
    const float* vr = vbase + (long)s*(2*DIM);
    o0 += aw*vr[d0]; o1 += aw*vr[d0+1]; o2 += aw*vr[d0+2];
  }
  float* orow = outp + (long)bl*DIM + head*96;
  orow[d0] = o0; orow[d0+1] = o1; orow[d0+2] = o2;
}

// ---------------- row log-softmax (768) + negative-entropy accumulation ----------------
__global__ void row_logsoftmax_kernel(const float* __restrict__ X, float* __restrict__ LP,
                                      float* __restrict__ ne_acc)
{
  __shared__ float red[8];
  int row = blockIdx.x;
  int b = row / SEQ;
  int t = threadIdx.x;                                 // 256 threads, 3 elems each
  const float* x = X + (long)row*DIM;
  float v0 = x[t], v1 = x[t+256], v2 = x[t+512];
  float m = fmaxf(v0, fmaxf(v1, v2));
  m = wave_max(m);
  if ((t & 31) == 0) red[t >> 5] = m;
  __syncthreads();
  float bm = red[0];
#pragma unroll
  for (int i = 1; i < 8; ++i) bm = fmaxf(bm, red[i]);
  __syncthreads();
  float e = expf(v0-bm) + expf(v1-bm) + expf(v2-bm);
  e = wave_sum(e);
  if ((t & 31) == 0) red[t >> 5] = e;
  __syncthreads();
  float bs = 0.f;
#pragma unroll
  for (int i = 0; i < 8; ++i) bs += red[i];
  __syncthreads();
  float lse = bm + logf(bs);
  float l0 = v0-lse, l1 = v1-lse, l2 = v2-lse;
  LP[(long)row*DIM + t      ] = l0;
  LP[(long)row*DIM + t + 256] = l1;
  LP[(long)row*DIM + t + 512] = l2;
  float ne = expf(l0)*l0 + expf(l1)*l1 + expf(l2)*l2;
  ne = wave_sum(ne);
  if ((t & 31) == 0) red[t >> 5] = ne;
  __syncthreads();
  if (t == 0) {
    float s = 0.f;
    for (int i = 0; i < 8; ++i) s += red[i];
    atomicAdd(ne_acc + b, s);
  }
}

// ---------------- column sums of p / log_p + fine_grained means ----------------
__global__ void col_stats_kernel(const float* __restrict__ lp_img, const float* __restrict__ lp_txt,
                                 const float* __restrict__ out_img, const float* __restrict__ out_txt,
                                 float* __restrict__ Sp_i, float* __restrict__ Slp_i,
                                 float* __restrict__ Sp_t, float* __restrict__ Slp_t,
                                 float* __restrict__ fine)
{
  int idx = blockIdx.x * 256 + threadIdx.x;            // 2*BATCH*DIM
  if (idx >= 2 * BATCH * DIM) return;
  int which = idx / (BATCH * DIM);
  int r = idx % (BATCH * DIM);
  int b = r / DIM, d = r % DIM;
  const float* LP = which ? lp_txt  : lp_img;
  const float* O  = which ? out_txt : out_img;
  const float* lpc = LP + ((long)b*SEQ)*DIM + d;
  const float* oc  = O  + ((long)b*SEQ)*DIM + d;
  float slp = 0.f, sp = 0.f, so = 0.f;
  for (int l = 0; l < SEQ; ++l) {
    float lp = lpc[(long)l*DIM];
    slp += lp; sp += expf(lp); so += oc[(long)l*DIM];
  }
  if (which) { Sp_t[r] = sp; Slp_t[r] = slp; } else { Sp_i[r] = sp; Slp_i[r] = slp; }
  fine[(long)b*(2*DIM) + which*DIM + d] = so * (1.0f/SEQ);
}

// ---------------- mask sums for sparsity ----------------
__global__ void mask_sum_kernel(const float* __restrict__ mask, float* __restrict__ acc)
{
  __shared__ float red[8];
  int b = blockIdx.y;
  const float* m = mask + (long)b * (SEQ*SEQ);
  float s = 0.f;
  for (int i = blockIdx.x*256 + threadIdx.x; i < SEQ*SEQ; i += 5*256) s += m[i];
  s = wave_sum(s);
  if ((threadIdx.x & 31) == 0) red[threadIdx.x >> 5] = s;
  __syncthreads();
  if (threadIdx.x == 0) {
    float t = 0.f;
    for (int i = 0; i < 8; ++i) t += red[i];
    atomicAdd(acc + b, t);
  }
}

// ---------------- finalize: KL + sparsities ----------------
// acc: [0..3]=Sne_img, [4..7]=Sne_txt, [8..11]=sum(mask_img), [12..15]=sum(mask_txt)
__global__ void finalize_kernel(const float* __restrict__ Sp_i, const float* __restrict__ Slp_i,
                                const float* __restrict__ Sp_t, const float* __restrict__ Slp_t,
                                const float* __restrict__ acc, float* __restrict__ out)
{
  __shared__ float red[8];
  int b = blockIdx.x, t = threadIdx.x;
  float dsum = 0.f;
  for (int d = t; d < DIM; d += 256)
    dsum += Sp_t[b*DIM+d]*Slp_i[b*DIM+d] + Sp_i[b*DIM+d]*Slp_t[b*DIM+d];
  dsum = wave_sum(dsum);
  if ((t & 31) == 0) red[t >> 5] = dsum;
  __syncthreads();
  if (t == 0) {
    float D = 0.f;
    for (int i = 0; i < 8; ++i) D += red[i];
    // binary mask => jm all-true => KL double sum factorizes
    float kl = 0.5f * ((float)SEQ * (acc[4+b] + acc[b]) - D) / (TOT_F + 1e-8f);
    out[b]        = kl;                    // kl_loss
    out[6148 + b] = acc[8+b]  / TOT_F;     // img_sparsity
    out[6152 + b] = acc[12+b] / TOT_F;     // txt_sparsity
  }
}

// ---------------- host ----------------
extern "C" void kernel_launch(void* const* d_in, const int* in_sizes, int n_in,
                              void* d_out, int out_size, void* d_ws, size_t ws_size,
                              hipStream_t stream)
{
  const float* img_seq = (const float*)d_in[0];
  const float* txt_seq = (const float*)d_in[1];
  const float* qW  = (const float*)d_in[2];
  const float* qb  = (const float*)d_in[3];
  const float* kvW = (const float*)d_in[4];
  const float* kvb = (const float*)d_in[5];
  const float* im_W1 = (const float*)d_in[6],  *im_b1 = (const float*)d_in[7];
  const float* im_gamma = (const float*)d_in[8], *im_beta = (const float*)d_in[9];
  const float* im_W2 = (const float*)d_in[10], *im_b2 = (const float*)d_in[11];
  const float* tm_W1 = (const float*)d_in[12], *tm_b1 = (const float*)d_in[13];
  const float* tm_gamma = (const float*)d_in[14], *tm_beta = (const float*)d_in[15];
  const float* tm_W2 = (const float*)d_in[16], *tm_b2 = (const float*)d_in[17];
  const float* g_img = (const float*)d_in[18];
  const float* g_txt = (const float*)d_in[19];
  float* out = (float*)d_out;

  // workspace layout (floats); total ~40.5 MB
  float* ws = (float*)d_ws;
  size_t o = 0;
  auto alloc = [&](size_t n){ float* p = ws + o; o += n; return p; };
  float* X_img  = alloc((size_t)ROWS*1536);
  float* X_txt  = alloc((size_t)ROWS*1536);
  float* Wa_im  = alloc((size_t)HID*1536);
  float* Wb_im  = alloc((size_t)HID*1536);
  float* Wa_tm  = alloc((size_t)HID*1536);
  float* Wb_tm  = alloc((size_t)HID*1536);
  float* a_im   = alloc((size_t)ROWS*HID);
  float* bt_im  = alloc((size_t)ROWS*HID);
  float* a_tm   = alloc((size_t)ROWS*HID);
  float* bt_tm  = alloc((size_t)ROWS*HID);
  float* q_img  = alloc((size_t)ROWS*DIM);
  float* q_txt  = alloc((size_t)ROWS*DIM);
  float* kv_img = alloc((size_t)ROWS*2*DIM);
  float* kv_txt = alloc((size_t)ROWS*2*DIM);
  float* sim    = alloc((size_t)BATCH*SEQ*SEQ);
  float* mask_i = alloc((size_t)BATCH*SEQ*SEQ);
  float* mask_t = alloc((size_t)BATCH*SEQ*SEQ);
  float* lp_img = alloc((size_t)ROWS*DIM);
  float* lp_txt = alloc((size_t)ROWS*DIM);
  float* Sp_i   = alloc((size_t)BATCH*DIM);
  float* Slp_i  = alloc((size_t)BATCH*DIM);
  float* Sp_t   = alloc((size_t)BATCH*DIM);
  float* Slp_t  = alloc((size_t)BATCH*DIM);
  float* gimg   = alloc((size_t)BATCH*DIM);
  float* gtxt   = alloc((size_t)BATCH*DIM);
  float* zrow   = alloc(KMAX);
  float* acc    = alloc(16);

  float* img_out = out + 6156;      // 4 + 4*1536 + 4 + 4
  float* txt_out = out + 611340;    // + 4*197*768

  auto gemm = [&](int nw, const float* A, const float* Bw, const float* bias, float* C,
                  int M, int N, int K, int lda, int ldb, int ldc,
                  long bsA, long bsB, long bsC, int nb, float alpha){
    int tM = (M + 15) / 16, tN = (N + 16*nw - 1) / (16*nw);
    dim3 grid((tM*tN + 7) / 8, nb);
    if (nw == 2)
      gemm_bt_wmma<2><<<grid, 256, 0, stream>>>(A, Bw, bias, C, zrow, M, N, K,
                                                lda, ldb, ldc, bsA, bsB, bsC, tM, tN, alpha);
    else
      gemm_bt_wmma<1><<<grid, 256, 0, stream>>>(A, Bw, bias, C, zrow, M, N, K,
                                                lda, ldb, ldc, bsA, bsB, bsC, tM, tN, alpha);
  };

  init_ws_kernel<<<1, 256, 0, stream>>>(zrow, acc);
  globals_kernel<<<(2*BATCH*DIM + 255)/256, 256, 0, stream>>>(img_seq, txt_seq, gimg, gtxt);
  build_x_kernel<<<((ROWS*1536) + 255)/256, 256, 0, stream>>>(img_seq, gimg, X_img);
  build_x_kernel<<<((ROWS*1536) + 255)/256, 256, 0, stream>>>(txt_seq, gtxt, X_txt);
  build_w_kernel<<<((HID*1536) + 255)/256, 256, 0, stream>>>(im_W1, Wa_im, Wb_im);
  build_w_kernel<<<((HID*1536) + 255)/256, 256, 0, stream>>>(tm_W1, Wa_tm, Wb_tm);

  // masker pre-activations (b1 folded into the 'a' GEMM bias)
  gemm(2, X_img, Wa_im, im_b1,   a_im,  ROWS, HID, 1536, 1536, 1536, HID, 0,0,0, 1, 1.f);
  gemm(2, X_txt, Wb_im, nullptr, bt_im, ROWS, HID, 1536, 1536, 1536, HID, 0,0,0, 1, 1.f);
  gemm(2, X_txt, Wa_tm, tm_b1,   a_tm,  ROWS, HID, 1536, 1536, 1536, HID, 0,0,0, 1, 1.f);
  gemm(2, X_img, Wb_tm, nullptr, bt_tm, ROWS, HID, 1536, 1536, 1536, HID, 0,0,0, 1, 1.f);
  // q / kv projections
  gemm(2, img_seq, qW,  qb,  q_img,  ROWS, DIM,   DIM, DIM, DIM, DIM,   0,0,0, 1, 1.f);
  gemm(2, txt_seq, qW,  qb,  q_txt,  ROWS, DIM,   DIM, DIM, DIM, DIM,   0,0,0, 1, 1.f);
  gemm(2, img_seq, kvW, kvb, kv_img, ROWS, 2*DIM, DIM, DIM, DIM, 2*DIM, 0,0,0, 1, 1.f);
  gemm(2, txt_seq, kvW, kvb, kv_txt, ROWS, 2*DIM, DIM, DIM, DIM, 2*DIM, 0,0,0, 1, 1.f);
  // sim[b] = img_seq[b] @ txt_seq[b]^T / sqrt(768); txt masker uses its transpose
  gemm(1, img_seq, txt_seq, nullptr, sim, SEQ, SEQ, DIM, DIM, DIM, SEQ,
       (long)SEQ*DIM, (long)SEQ*DIM, (long)SEQ*SEQ, BATCH, 0.03608439182435161f);

  masker_kernel<<<(PAIRS + 7)/8, 256, 0, stream>>>(a_im, bt_im, im_gamma, im_beta,
                                                   im_W2, im_b2, sim, g_img, mask_i, 0);
  masker_kernel<<<(PAIRS + 7)/8, 256, 0, stream>>>(a_tm, bt_tm, tm_gamma, tm_beta,
                                                   tm_W2, tm_b2, sim, g_txt, mask_t, 1);

  attention_kernel<<<ROWS, 256, 0, stream>>>(q_img, kv_txt, mask_i, img_out);
  attention_kernel<<<ROWS, 256, 0, stream>>>(q_txt, kv_img, mask_t, txt_out);

  row_logsoftmax_kernel<<<ROWS, 256, 0, stream>>>(img_out, lp_img, acc + 0);
  row_logsoftmax_kernel<<<ROWS, 256, 0, stream>>>(txt_out, lp_txt, acc + 4);

  col_stats_kernel<<<(2*BATCH*DIM + 255)/256, 256, 0, stream>>>(
      lp_img, lp_txt, img_out, txt_out, Sp_i, Slp_i, Sp_t, Slp_t, out + 4);

  mask_sum_kernel<<<dim3(5, BATCH), 256, 0, stream>>>(mask_i, acc + 8);
  mask_sum_kernel<<<dim3(5, BATCH), 256, 0, stream>>>(mask_t, acc + 12);

  finalize_kernel<<<BATCH, 256, 0, stream>>>(Sp_i, Slp_i, Sp_t, Slp_t, acc, out);
}